// SMG_mulithead_3942779977729
// MI455X (gfx1250) — compile-verified
//
#include <hip/hip_runtime.h>
#include <math.h>

#define N_NODES 100000
#define N_EDGES 1600000
#define FEATS   128
#define KHOP    3
#define NHEAD   3

typedef float v2f __attribute__((ext_vector_type(2)));
typedef float v8f __attribute__((ext_vector_type(8)));

// Native, non-returning f32 atomic add (GLOBAL_ATOMIC_ADD_F32, STOREcnt path).
__device__ __forceinline__ void atomic_add_f32(float* p, float v) {
    __hip_atomic_fetch_add(p, v, __ATOMIC_RELAXED, __HIP_MEMORY_SCOPE_AGENT);
}

// ---------------------------------------------------------------------------
// 1. Degree accumulation: degs[dst[e]] += 1
// ---------------------------------------------------------------------------
__global__ void __launch_bounds__(256)
smg_deg_kernel(const int* __restrict__ dst, float* __restrict__ degs) {
    int e = blockIdx.x * blockDim.x + threadIdx.x;
    if (e < N_EDGES) atomic_add_f32(&degs[dst[e]], 1.0f);
}

// ---------------------------------------------------------------------------
// 2. norm[n] = clip(deg,1)^-0.5   (in place over the degree buffer)
// ---------------------------------------------------------------------------
__global__ void __launch_bounds__(256)
smg_norm_kernel(float* __restrict__ degs) {
    int n = blockIdx.x * blockDim.x + threadIdx.x;
    if (n < N_NODES) {
        float d = fmaxf(degs[n], 1.0f);
        degs[n] = 1.0f / sqrtf(d);
    }
}

// ---------------------------------------------------------------------------
// 3. One propagation hop: hout[dst] += norm[src]*norm[dst] * hin[src]
//    One wave per edge; each lane moves 4 contiguous floats (float4 load,
//    4 native global f32 atomic adds). Both norm factors folded per edge.
// ---------------------------------------------------------------------------
__global__ void __launch_bounds__(256)
smg_propagate_kernel(const float* __restrict__ hin,
                     float* __restrict__ hout,
                     const int* __restrict__ src,
                     const int* __restrict__ dst,
                     const float* __restrict__ norm) {
    int wid  = (blockIdx.x * blockDim.x + threadIdx.x) >> 5;
    int lane = threadIdx.x & 31;
    if (wid >= N_EDGES) return;
    int s = src[wid];
    int d = dst[wid];
    float scale = norm[s] * norm[d];
    const float4 v = *(const float4*)(hin + (size_t)s * FEATS + lane * 4);
    float* o = hout + (size_t)d * FEATS + lane * 4;
    atomic_add_f32(o + 0, v.x * scale);
    atomic_add_f32(o + 1, v.y * scale);
    atomic_add_f32(o + 2, v.z * scale);
    atomic_add_f32(o + 3, v.w * scale);
}

// ---------------------------------------------------------------------------
// 4. Per-node head logits -> log_softmax over k -> entropy + combination
//    weights w[n][k] = alpha[k] * sum_h p[h][n][k].
//    One wave per node; lane L covers features [4L, 4L+4); 12 dot products
//    accumulated per lane then reduced with xor-shuffles.
// ---------------------------------------------------------------------------
__global__ void __launch_bounds__(256)
smg_weights_kernel(const float* __restrict__ feat,
                   const float* __restrict__ h1,
                   const float* __restrict__ h2,
                   const float* __restrict__ h3,
                   const float* __restrict__ head_w,
                   const float* __restrict__ head_b,
                   const float* __restrict__ alpha,
                   float* __restrict__ wbuf,
                   float* __restrict__ entropy_out) {
    int wid  = (blockIdx.x * blockDim.x + threadIdx.x) >> 5;
    int lane = threadIdx.x & 31;
    if (wid >= N_NODES) return;

    const float* rows[KHOP + 1] = {
        feat + (size_t)wid * FEATS,
        h1   + (size_t)wid * FEATS,
        h2   + (size_t)wid * FEATS,
        h3   + (size_t)wid * FEATS
    };

    float4 hw[NHEAD];
#pragma unroll
    for (int h = 0; h < NHEAD; ++h)
        hw[h] = *(const float4*)(head_w + (size_t)h * FEATS + lane * 4);

    float acc[NHEAD][KHOP + 1];
#pragma unroll
    for (int h = 0; h < NHEAD; ++h)
#pragma unroll
        for (int k = 0; k <= KHOP; ++k) acc[h][k] = 0.0f;

#pragma unroll
    for (int k = 0; k <= KHOP; ++k) {
        float4 v = *(const float4*)(rows[k] + lane * 4);
#pragma unroll
        for (int h = 0; h < NHEAD; ++h)
            acc[h][k] += v.x * hw[h].x + v.y * hw[h].y + v.z * hw[h].z + v.w * hw[h].w;
    }

    // full-wave reduction (result valid in every lane)
#pragma unroll
    for (int h = 0; h < NHEAD; ++h)
#pragma unroll
        for (int k = 0; k <= KHOP; ++k)
#pragma unroll
            for (int off = 16; off > 0; off >>= 1)
                acc[h][k] += __shfl_xor(acc[h][k], off, 32);

    if (lane == 0) {
        float g[KHOP + 1] = {0.0f, 0.0f, 0.0f, 0.0f};
        float ent = 0.0f;
#pragma unroll
        for (int h = 0; h < NHEAD; ++h) {
            float l[KHOP + 1];
            float m = -3.4e38f;
#pragma unroll
            for (int k = 0; k <= KHOP; ++k) {
                l[k] = acc[h][k] + head_b[h];
                m = fmaxf(m, l[k]);
            }
            float s = 0.0f;
#pragma unroll
            for (int k = 0; k <= KHOP; ++k) s += expf(l[k] - m);
            float lse = logf(s);
#pragma unroll
            for (int k = 0; k <= KHOP; ++k) {
                float lp = l[k] - m - lse;
                float p  = expf(lp);
                g[k] += p;
                ent  -= p * lp;
            }
        }
        float4 w;
        w.x = alpha[0] * g[0];
        w.y = alpha[1] * g[1];
        w.z = alpha[2] * g[2];
        w.w = alpha[3] * g[3];
        *(float4*)(wbuf + (size_t)wid * 4) = w;
        entropy_out[wid] = ent;
    }
}

// ---------------------------------------------------------------------------
// 5. Final GEMM via V_WMMA_F32_16X16X4_F32 (exact fp32 matrix pipe):
//    result[n][j] = sum_f (sum_k w[n][k]*feats_k[n][f]) * fc_w[j][f] + 4*fc_b[j]
//    One wave computes one 16x16 output tile; the k-combination of the 4
//    feature hops is fused into the A-fragment build (4 FMAs / element).
//    A-frag layout (16x4 f32): lanes 0-15 hold K=0,1 of row M=lane,
//    lanes 16-31 hold K=2,3. B-frag mirrored with N=lane&15.
// ---------------------------------------------------------------------------
__global__ void __launch_bounds__(256)
smg_gemm_wmma_kernel(const float* __restrict__ feat,
                     const float* __restrict__ h1,
                     const float* __restrict__ h2,
                     const float* __restrict__ h3,
                     const float* __restrict__ wbuf,
                     const float* __restrict__ fcw,
                     const float* __restrict__ fcb,
                     float* __restrict__ out) {
    int wid  = (blockIdx.x * blockDim.x + threadIdx.x) >> 5;
    int lane = threadIdx.x & 31;
    int mtile = wid >> 3;        // 6250 M tiles (100000 = 6250*16, no tail)
    int ntile = wid & 7;         // 8 N tiles
    if (mtile >= N_NODES / 16) return;

    int half  = lane >> 4;       // 0 or 1
    int l15   = lane & 15;
    int row   = mtile * 16 + l15;    // A row owned by this lane
    int j     = ntile * 16 + l15;    // fc_w row (== output column)
    int khalf = half * 2;            // K sub-offset within each 4-wide step

    float4 wr = *(const float4*)(wbuf + (size_t)row * 4);
    const float* f0 = feat + (size_t)row * FEATS;
    const float* f1 = h1   + (size_t)row * FEATS;
    const float* f2 = h2   + (size_t)row * FEATS;
    const float* f3 = h3   + (size_t)row * FEATS;
    const float* bp = fcw  + (size_t)j   * FEATS;

    v8f c = {};
#pragma unroll 8
    for (int k0 = 0; k0 < FEATS; k0 += 4) {
        int kk = k0 + khalf;
        v2f x0 = *(const v2f*)(f0 + kk);
        v2f x1 = *(const v2f*)(f1 + kk);
        v2f x2 = *(const v2f*)(f2 + kk);
        v2f x3 = *(const v2f*)(f3 + kk);
        v2f a, b;
        a.x = wr.x * x0.x + wr.y * x1.x + wr.z * x2.x + wr.w * x3.x;
        a.y = wr.x * x0.y + wr.y * x1.y + wr.z * x2.y + wr.w * x3.y;
        b = *(const v2f*)(bp + kk);
        c = __builtin_amdgcn_wmma_f32_16x16x4_f32(
                /*neg_a=*/false, a, /*neg_b=*/false, b,
                /*c_mod=*/(short)0, c, /*reuse_a=*/false, /*reuse_b=*/false);
    }

    // D tile: VGPR r -> (M = r + 8*half, N = l15)
    float bias  = 4.0f * fcb[j];
    float* op   = out + (size_t)(mtile * 16 + half * 8) * FEATS + ntile * 16 + l15;
#pragma unroll
    for (int r = 0; r < 8; ++r)
        op[(size_t)r * FEATS] = c[r] + bias;
}

// ---------------------------------------------------------------------------
// Launch
// ---------------------------------------------------------------------------
extern "C" void kernel_launch(void* const* d_in, const int* in_sizes, int n_in,
                              void* d_out, int out_size, void* d_ws, size_t ws_size,
                              hipStream_t stream) {
    const float* feat   = (const float*)d_in[0];
    const float* fc_w   = (const float*)d_in[1];
    const float* fc_b   = (const float*)d_in[2];
    const float* alpha  = (const float*)d_in[3];
    const float* head_w = (const float*)d_in[4];
    const float* head_b = (const float*)d_in[5];
    const int*   src    = (const int*)d_in[6];
    const int*   dst    = (const int*)d_in[7];

    float* out     = (float*)d_out;
    float* entropy = out + (size_t)N_NODES * FEATS;

    const size_t NF = (size_t)N_NODES * FEATS;
    float* ws   = (float*)d_ws;
    float* h1   = ws;
    float* h2   = ws + NF;
    float* h3   = ws + 2 * NF;
    float* norm = ws + 3 * NF;            // degree buffer, then norm in place
    float* wbuf = norm + N_NODES;         // per-node [alpha_k * g_k], 4 floats

    // degrees -> norm
    hipMemsetAsync(norm, 0, N_NODES * sizeof(float), stream);
    smg_deg_kernel<<<(N_EDGES + 255) / 256, 256, 0, stream>>>(dst, norm);
    smg_norm_kernel<<<(N_NODES + 255) / 256, 256, 0, stream>>>(norm);

    // K propagation hops (wave per edge, 8 waves per block)
    float* houts[KHOP] = {h1, h2, h3};
    const float* hin = feat;
    for (int k = 0; k < KHOP; ++k) {
        hipMemsetAsync(houts[k], 0, NF * sizeof(float), stream);
        smg_propagate_kernel<<<N_EDGES / 8, 256, 0, stream>>>(hin, houts[k], src, dst, norm);
        hin = houts[k];
    }

    // per-node softmax weights + entropy (wave per node)
    smg_weights_kernel<<<N_NODES / 8, 256, 0, stream>>>(
        feat, h1, h2, h3, head_w, head_b, alpha, wbuf, entropy);

    // fused combine + GEMM on the fp32 WMMA pipe (wave per 16x16 tile)
    smg_gemm_wmma_kernel<<<(N_NODES / 16), 256, 0, stream>>>(
        feat, h1, h2, h3, wbuf, fc_w, fc_b, out);
}